// EquivariantMatrix_89429809038047
// MI455X (gfx1250) — compile-verified
//
#include <hip/hip_runtime.h>

// CDNA5 / gfx1250 — fp32 WMMA group-equivariant convolution with TDM staging.
// out[b,f,h] = sum_{i,g} x[b,i,g] * kernel[f,i,(h "minus" g)] + bias[f]
// Per h: C(16x16) = A(16 x 8192) * B(8192 x 16) via V_WMMA_F32_16X16X4_F32.
// Kernel/x slices are DMA'd into double-buffered LDS by the Tensor Data Mover
// (tensor_load_to_lds, TENSORcnt), with HW row padding producing the
// bank-conflict-free LDS strides. The group permutation pt[g,h] = h "minus" g is
// applied in LDS read addressing, so one staged slice serves all f-tiles and
// both h values of the block.

typedef float v2f __attribute__((ext_vector_type(2)));
typedef float v8f __attribute__((ext_vector_type(8)));
typedef unsigned int u32x4 __attribute__((ext_vector_type(4)));
typedef int i32x4 __attribute__((ext_vector_type(4)));
typedef int i32x8 __attribute__((ext_vector_type(8)));

constexpr int SYMM = 256;   // group size (16x16 lattice)
constexpr int NIN  = 32;    // in_features
constexpr int NF   = 64;    // features
constexpr int NB   = 16;    // batch
constexpr int KU_STRIDE = 257;               // LDS stride (floats) kernel slice [f][s]
constexpr int XS_STRIDE = 258;               // LDS stride (floats) x slice      [b][g]
constexpr int KU_FLOATS = NF * KU_STRIDE;    // 16448
constexpr int BUF_FLOATS = KU_FLOATS + NB * XS_STRIDE;  // 20576 (~80.4 KB)

#if defined(__HIP_DEVICE_COMPILE__) && __has_builtin(__builtin_amdgcn_tensor_load_to_lds)
#define USE_TDM 1
#else
#define USE_TDM 0
#endif

__device__ __forceinline__ unsigned lds_off(const void* p) {
    // Flat LDS pointers: addr[63:32] = SHARED_BASE aperture, addr[31:0] = LDS byte offset.
    return (unsigned)(unsigned long long)p;
}

#if USE_TDM
// 2D TDM tile: nrows rows of 256 contiguous fp32, global row stride 8192 elems,
// LDS destination padded by (pad_amt_enc+1) dwords after every 256 dwords.
__device__ __forceinline__ void tdm_load_2d(const void* gaddr, unsigned lds_byte,
                                            int nrows, int pad_amt_enc) {
    unsigned long long ga = (unsigned long long)gaddr;
    u32x4 g0;
    g0.x = 1u;                                              // count=1, user desc
    g0.y = lds_byte;                                        // lds_addr [63:32]
    g0.z = (unsigned)(ga & 0xffffffffu);                    // global_addr [95:64]
    g0.w = (unsigned)((ga >> 32) & 0x01ffffffu) | (1u << 31); // addr[56:32] | type=2
    i32x8 g1;
    g1[0] = (2 << 16) | (1 << 20) | (7 << 22) | (pad_amt_enc << 25);
            // data_size=4B | pad_enable | pad_interval=256dw | pad_amount
    g1[1] = (int)(256u << 16);                 // tensor_dim0 = 256
    g1[2] = (int)((unsigned)nrows << 16);      // tensor_dim1 = nrows
    g1[3] = (int)(256u << 16);                 // tile_dim0 = 256
    g1[4] = nrows;                             // tile_dim1 = nrows, tile_dim2 = 0
    g1[5] = NIN * SYMM;                        // tensor_dim0_stride = 8192 elems
    g1[6] = 0;
    g1[7] = 0;
    i32x4 z4 = {0, 0, 0, 0};
#if defined(__clang_major__) && __clang_major__ >= 23
    i32x8 z8 = {0, 0, 0, 0, 0, 0, 0, 0};
    __builtin_amdgcn_tensor_load_to_lds(g0, g1, z4, z4, z8, 0);
#else
    __builtin_amdgcn_tensor_load_to_lds(g0, g1, z4, z4, 0);
#endif
}
#endif

__global__ __launch_bounds__(256, 1)
void equi_wmma_f32(const float* __restrict__ x,
                   const float* __restrict__ kern,
                   const float* __restrict__ bias,
                   float* __restrict__ out)
{
    extern __shared__ float smem[];

    const int tid      = threadIdx.x;      // 0..255 (8 waves)
    const int lane     = tid & 31;
    const int wave     = tid >> 5;         // 0..7
    const int ftile    = wave & 3;         // 16-feature tile
    const int hloc     = wave >> 2;        // 0..1
    const int h        = blockIdx.x * 2 + hloc;
    const int l16      = lane & 15;
    const int laneHalf = lane >> 4;        // 0: K rows {0,1}, 1: K rows {2,3}

    const int hc = h >> 4, hd = h & 15;    // lattice coords of h

    v8f acc0 = {}, acc1 = {}, acc2 = {}, acc3 = {};

#if USE_TDM
    if (tid == 0) {                         // prologue: DMA slice 0 into buffer 0
        tdm_load_2d(kern, lds_off(smem), NF, 0);
        tdm_load_2d(x,    lds_off(smem + KU_FLOATS), NB, 1);
    }
#endif

    for (int i = 0; i < NIN; ++i) {
#if USE_TDM
        float* buf = smem + (i & 1) * BUF_FLOATS;
        __syncthreads();   // all waves done reading buf[(i+1)&1] (iter i-1 compute)
        if (tid == 0) {
            __builtin_amdgcn_s_wait_tensorcnt(0);    // slice i landed in buf
            if (i + 1 < NIN) {                       // DMA slice i+1, overlaps compute
                float* nb = smem + ((i + 1) & 1) * BUF_FLOATS;
                tdm_load_2d(kern + (i + 1) * SYMM, lds_off(nb), NF, 0);
                tdm_load_2d(x    + (i + 1) * SYMM, lds_off(nb + KU_FLOATS), NB, 1);
            }
        }
        __syncthreads();   // buf visible to all waves
#else
        float* buf = smem;
        __syncthreads();
        for (int j = tid; j < NF * SYMM; j += 256) {
            int f = j >> 8, s = j & 255;
            buf[f * KU_STRIDE + s] = kern[(f * NIN + i) * SYMM + s];
        }
        for (int j = tid; j < NB * SYMM; j += 256) {
            int b = j >> 8, g = j & 255;
            buf[KU_FLOATS + b * XS_STRIDE + g] = x[(b * NIN + i) * SYMM + g];
        }
        __syncthreads();
#endif
        const float* Ku = buf;                       // [NF][KU_STRIDE]
        const float* Xs = buf + KU_FLOATS;           // [NB][XS_STRIDE]
        const float* xrow = Xs + l16 * XS_STRIDE + laneHalf * 2;
        const float* kcol = Ku + (ftile * 16 + l16) * KU_STRIDE;

        auto step = [&](int c, v8f acc) -> v8f {
            const int g0 = c + laneHalf * 2;
            const int g1 = g0 + 1;
            // s = pt[g,h] = ((hc-gc)&15)*16 + ((hd-gd)&15)
            const int s0 = (((hc - (g0 >> 4)) & 15) << 4) | ((hd - (g0 & 15)) & 15);
            const int s1 = (((hc - (g1 >> 4)) & 15) << 4) | ((hd - (g1 & 15)) & 15);
            v2f a = *(const v2f*)(xrow + c);         // A[M=l16][K=g0,g1]
            v2f bfrag;
            bfrag.x = kcol[s0];                      // B[K=g0][N]
            bfrag.y = kcol[s1];                      // B[K=g1][N]
            return __builtin_amdgcn_wmma_f32_16x16x4_f32(
                false, a, false, bfrag, (short)0, acc, false, false);
        };

        #pragma unroll
        for (int gc = 0; gc < SYMM; gc += 16) {      // 4 independent acc chains
            acc0 = step(gc + 0,  acc0);
            acc1 = step(gc + 4,  acc1);
            acc2 = step(gc + 8,  acc2);
            acc3 = step(gc + 12, acc3);
        }
    }

    // epilogue: C layout (ISA 7.12.2): vgpr r -> M = r + laneHalf*8, N = l16
    const int f  = ftile * 16 + l16;
    const float bv = bias[f];
    #pragma unroll
    for (int r = 0; r < 8; ++r) {
        const int b = r + laneHalf * 8;
        const float v = acc0[r] + acc1[r] + acc2[r] + acc3[r] + bv;
        out[(b * NF + f) * SYMM + h] = v;
    }
}

extern "C" void kernel_launch(void* const* d_in, const int* in_sizes, int n_in,
                              void* d_out, int out_size, void* d_ws, size_t ws_size,
                              hipStream_t stream) {
    (void)in_sizes; (void)n_in; (void)d_ws; (void)ws_size; (void)out_size;
    const float* x    = (const float*)d_in[0];   // (16, 32, 256) f32
    const float* kern = (const float*)d_in[1];   // (64, 32, 256) f32
    const float* bias = (const float*)d_in[2];   // (64,) f32
    // d_in[3] = product_table (256,256) i32 — reproduced arithmetically on-device.
    float* out = (float*)d_out;                  // (16, 64, 256) f32

    const size_t smemBytes = (size_t)(2 * BUF_FLOATS) * sizeof(float);  // ~161 KB
    dim3 grid(SYMM / 2);   // 128 blocks, 2 h-values each
    dim3 block(256);       // 8 waves: 4 f-tiles x 2 h-values
    equi_wmma_f32<<<grid, block, smemBytes, stream>>>(x, kern, bias, out);
}